// DiscreteExactLoss_57071525430111
// MI455X (gfx1250) — compile-verified
//
#include <hip/hip_runtime.h>
#include <hip/hip_bf16.h>
#include <stdint.h>

typedef float v2f __attribute__((ext_vector_type(2)));
typedef float v8f __attribute__((ext_vector_type(8)));

// ---------------------------------------------------------------------------
// Kernel 0: zero the 1024-float accumulator matrix M (32x32) in workspace.
// ---------------------------------------------------------------------------
__global__ void del_zero_kernel(float* __restrict__ M) {
    M[blockIdx.x * 256 + threadIdx.x] = 0.0f;
}

// ---------------------------------------------------------------------------
// Kernel 1: accumulate M[i][j] = sum_b u_b[i] * v_b[j] via WMMA f32 16x16x4.
//   u_b[i] = prod_{r=0..4} sel(act[b][r],   bit_{4-r}(i))
//   v_b[j] = prod_{r=5..9} sel(act[b][r], bit_{4-(r-5)}(j))
//   sel(a, 0) = 1-a ; sel(a, 1) = a.
// Staging: 32 samples x 10 floats per block iteration, copied straight into
// LDS with global_load_async_to_lds_b32 (ASYNCcnt), no VGPR bounce.
// Each wave holds a 32x32 f32 accumulator (4 WMMA tiles), K-step = 4 samples.
// ---------------------------------------------------------------------------
__global__ __launch_bounds__(256) void del_accum_kernel(
    const float* __restrict__ act, float* __restrict__ M,
    int B, int samplesPerBlock) {

    __shared__ float sAct[320];   // 32 samples x 10 floats, unpadded

    const int t    = threadIdx.x;
    const int lane = t & 31;
    const int wave = t >> 5;                 // 0..7
    const int half = (lane >> 4) & 1;        // 0 -> K slots 0,1 ; 1 -> K 2,3
    const int li   = lane & 15;              // low 4 bits of row/col index
    const int b1 = (li >> 3) & 1;
    const int b2 = (li >> 2) & 1;
    const int b3 = (li >> 1) & 1;
    const int b4 =  li       & 1;

    const int total      = B * 10;
    const int blockStart = blockIdx.x * samplesPerBlock;

    // Wave-relative LDS byte offset of sAct: low 32 bits of the generic
    // pointer (LDS aperture occupies the high dword per the ISA flat rules).
    const uint32_t ldsBase = (uint32_t)(uintptr_t)(&sAct[0]);

    v8f a00 = {}, a01 = {}, a10 = {}, a11 = {};

    for (int it = 0; it < samplesPerBlock; it += 32) {
        const int s0 = blockStart + it;

        if (s0 + 32 <= B) {
            // Fast path: async-copy 320 contiguous floats global -> LDS.
            const uint64_t gbase = (uint64_t)(uintptr_t)(act + (size_t)s0 * 10);
            asm volatile("global_load_async_to_lds_b32 %0, %1, off"
                         :: "v"(ldsBase + (uint32_t)(t * 4)),
                            "v"(gbase + (uint64_t)(t * 4))
                         : "memory");
            if (t < 64) {
                asm volatile("global_load_async_to_lds_b32 %0, %1, off"
                             :: "v"(ldsBase + (uint32_t)((t + 256) * 4)),
                                "v"(gbase + (uint64_t)((t + 256) * 4))
                             : "memory");
            }
            asm volatile("s_wait_asynccnt 0" ::: "memory");
        } else {
            // Tail path (not taken for B = 131072): guarded scalar staging.
            for (int e = t; e < 320; e += 256) {
                const int g = s0 * 10 + e;
                sAct[e] = (g < total) ? act[g] : 0.0f;
            }
        }
        __syncthreads();

        // This wave consumes local samples wave*4 .. wave*4+3.
        // This lane supplies K slots (half*2, half*2+1) at row/col index li.
        const int ls = wave * 4 + half * 2;
        v2f A0, A1, B0, B1;
#pragma unroll
        for (int q = 0; q < 2; ++q) {
            const int   s    = ls + q;
            const float mask = ((s0 + s) < B) ? 1.0f : 0.0f;
            const float r0 = sAct[s * 10 + 0], r1 = sAct[s * 10 + 1];
            const float r2 = sAct[s * 10 + 2], r3 = sAct[s * 10 + 3];
            const float r4 = sAct[s * 10 + 4], r5 = sAct[s * 10 + 5];
            const float r6 = sAct[s * 10 + 6], r7 = sAct[s * 10 + 7];
            const float r8 = sAct[s * 10 + 8], r9 = sAct[s * 10 + 9];

            const float pu = (b1 ? r1 : 1.0f - r1) * (b2 ? r2 : 1.0f - r2)
                           * (b3 ? r3 : 1.0f - r3) * (b4 ? r4 : 1.0f - r4)
                           * mask;                      // mask once (A side)
            const float pv = (b1 ? r6 : 1.0f - r6) * (b2 ? r7 : 1.0f - r7)
                           * (b3 ? r8 : 1.0f - r8) * (b4 ? r9 : 1.0f - r9);

            A0[q] = (1.0f - r0) * pu;   // rows  0..15 (reg0 bit = 0)
            A1[q] =         r0  * pu;   // rows 16..31 (reg0 bit = 1)
            B0[q] = (1.0f - r5) * pv;   // cols  0..15
            B1[q] =         r5  * pv;   // cols 16..31
        }

        // D = A(16x4) x B(4x16) + C, f32 in / f32 accumulate.
        a00 = __builtin_amdgcn_wmma_f32_16x16x4_f32(false, A0, false, B0,
                                                    (short)0, a00, false, false);
        a01 = __builtin_amdgcn_wmma_f32_16x16x4_f32(false, A0, false, B1,
                                                    (short)0, a01, false, false);
        a10 = __builtin_amdgcn_wmma_f32_16x16x4_f32(false, A1, false, B0,
                                                    (short)0, a10, false, false);
        a11 = __builtin_amdgcn_wmma_f32_16x16x4_f32(false, A1, false, B1,
                                                    (short)0, a11, false, false);

        __syncthreads();
    }

    // C/D layout: VGPR p -> M = p + 8*half, N = li.
    const int mAdd = half * 8;
#pragma unroll
    for (int p = 0; p < 8; ++p) {
        atomicAdd(&M[(p + mAdd)      * 32 + li     ], a00[p]);
        atomicAdd(&M[(p + mAdd)      * 32 + li + 16], a01[p]);
        atomicAdd(&M[(p + mAdd + 16) * 32 + li     ], a10[p]);
        atomicAdd(&M[(p + mAdd + 16) * 32 + li + 16], a11[p]);
    }
}

// ---------------------------------------------------------------------------
// Kernel 2: out = sum_e p_e * log2(max(p_e, 1e-12)), p_e = M[e] / B.
// (Reference returns -joint_h = sum p*log2(clip(p)).)
// ---------------------------------------------------------------------------
__global__ __launch_bounds__(256) void del_entropy_kernel(
    const float* __restrict__ M, float* __restrict__ out, int B) {

    __shared__ float red[256];
    const float invB = 1.0f / (float)B;
    float s = 0.0f;
    for (int e = threadIdx.x; e < 1024; e += 256) {
        const float p  = M[e] * invB;
        const float ps = fmaxf(p, 1e-12f);
        s += p * __log2f(ps);
    }
    red[threadIdx.x] = s;
    __syncthreads();
    for (int off = 128; off > 0; off >>= 1) {
        if (threadIdx.x < off) red[threadIdx.x] += red[threadIdx.x + off];
        __syncthreads();
    }
    if (threadIdx.x == 0) out[0] = red[0];
}

// ---------------------------------------------------------------------------
extern "C" void kernel_launch(void* const* d_in, const int* in_sizes, int n_in,
                              void* d_out, int out_size, void* d_ws, size_t ws_size,
                              hipStream_t stream) {
    const float* act = (const float*)d_in[0];
    const int total  = in_sizes[0];
    const int B      = total / 10;           // 131072 samples, R = 10
    float* M         = (float*)d_ws;         // 1024 f32 accumulator

    del_zero_kernel<<<4, 256, 0, stream>>>(M);

    const int nBlocks = 64;
    int spb = (B + nBlocks - 1) / nBlocks;   // samples per block
    spb = (spb + 31) & ~31;                  // multiple of 32 (8 waves x 4)

    del_accum_kernel<<<nBlocks, 256, 0, stream>>>(act, M, B, spb);
    del_entropy_kernel<<<1, 256, 0, stream>>>(M, (float*)d_out, B);
}